// DenseCRF_62801011802605
// MI455X (gfx1250) — compile-verified
//
#include <hip/hip_runtime.h>
#include <math.h>

#define B_  8
#define C_  21
#define H_  512
#define W_  512
#define HW_ (H_*W_)

typedef __attribute__((ext_vector_type(2))) float v2f;
typedef __attribute__((ext_vector_type(8))) float v8f;

__device__ __forceinline__ int reflect_idx(int i, int n) {
  if (i < 0) i = -i;
  if (i >= n) i = 2*n - 2 - i;
  return i;
}

// ---------------------------------------------------------------------------
// Q0 = softmax(unary, axis=class). One thread per pixel, classes strided HW.
// ---------------------------------------------------------------------------
__global__ __launch_bounds__(256) void crf_softmax_init(
    const float* __restrict__ unary, float* __restrict__ q) {
  const int p   = blockIdx.x * 256 + threadIdx.x;     // 0 .. B*HW-1 (exact)
  const int b   = p / HW_;
  const int pin = p - b * HW_;
  const float* u = unary + (size_t)b * C_ * HW_ + pin;
  float* o       = q     + (size_t)b * C_ * HW_ + pin;
  float v[C_];
  float m = -3.4e38f;
#pragma unroll
  for (int i = 0; i < C_; ++i) { v[i] = u[(size_t)i * HW_]; m = fmaxf(m, v[i]); }
  float s = 0.f;
#pragma unroll
  for (int i = 0; i < C_; ++i) { v[i] = __expf(v[i] - m); s += v[i]; }
  const float inv = 1.f / s;
#pragma unroll
  for (int i = 0; i < C_; ++i) o[(size_t)i * HW_] = v[i] * inv;
}

// ---------------------------------------------------------------------------
// Separable 7x7 Gaussian blur with reflect padding, fused in LDS.
// Qs = blur(Q) * (pos_w + bi_w).  One block = 32x32 tile of one (b,c) plane.
// ---------------------------------------------------------------------------
__global__ __launch_bounds__(256) void crf_blur(
    const float* __restrict__ q, float* __restrict__ qs,
    const float* __restrict__ pos_w, const float* __restrict__ bi_w) {
  __shared__ float sIn[38][40];
  __shared__ float sH[38][32];

  float g[7];
  {
    float ssum = 0.f;
#pragma unroll
    for (int i = 0; i < 7; ++i) {
      const float d = (float)(i - 3);
      g[i] = __expf(-d * d * 0.125f);        // sigma = 2 -> 1/(2*sigma^2) = 0.125
      ssum += g[i];
    }
    const float rs = 1.f / ssum;
#pragma unroll
    for (int i = 0; i < 7; ++i) g[i] *= rs;
  }

  const int plane = blockIdx.z;              // b*C + c
  const float* src = q  + (size_t)plane * HW_;
  float*       dst = qs + (size_t)plane * HW_;
  const int w0 = blockIdx.x * 32, h0 = blockIdx.y * 32;
  const int tid = threadIdx.x;

  for (int i = tid; i < 38 * 38; i += 256) {
    const int r = i / 38, c = i - r * 38;
    const int gh = reflect_idx(h0 + r - 3, H_);
    const int gw = reflect_idx(w0 + c - 3, W_);
    sIn[r][c] = src[gh * W_ + gw];
  }
  __syncthreads();

  for (int i = tid; i < 38 * 32; i += 256) {
    const int r = i >> 5, c = i & 31;
    float a = 0.f;
#pragma unroll
    for (int k = 0; k < 7; ++k) a = fmaf(g[k], sIn[r][c + k], a);
    sH[r][c] = a;
  }
  __syncthreads();

  const float scale = pos_w[0] + bi_w[0];
#pragma unroll
  for (int ii = 0; ii < 4; ++ii) {
    const int i = tid + ii * 256;            // 1024 outputs, exact
    const int r = i >> 5, c = i & 31;
    float a = 0.f;
#pragma unroll
    for (int k = 0; k < 7; ++k) a = fmaf(g[k], sH[r + k][c], a);
    dst[(size_t)(h0 + r) * W_ + (w0 + c)] = a * scale;
  }
}

// ---------------------------------------------------------------------------
// Q <- softmax(Q - compat @ Qs, axis=class), in place.
// One wave handles 16 pixels. compat padded to 32x24; D = A(16x4) x B(4x16) + C
// via V_WMMA_F32_16X16X4_F32, 2 M-tiles x 6 K-steps.
// ---------------------------------------------------------------------------
__global__ __launch_bounds__(256) void crf_compat_softmax(
    float* __restrict__ q, const float* __restrict__ qs,
    const float* __restrict__ compat) {
  const int lane = threadIdx.x & 31;
  const bool hi  = lane >= 16;
  const int  ln  = lane & 15;
  const int wave = (blockIdx.x * 256 + threadIdx.x) >> 5;
  const int P0   = wave * 16;                // first pixel of this wave's group
  const int b    = P0 / HW_;                 // 16-groups never cross batches
  const int pin  = P0 - b * HW_;
  float*       qB  = q  + (size_t)b * C_ * HW_;
  const float* qsB = qs + (size_t)b * C_ * HW_;
  const int pix = pin + ln;

  // --- Preload A (compat, zero-padded to 32 rows x 24 cols) ---
  // A 16x4 f32 layout: lanes 0-15 M=0..15 {K0,K1}; lanes 16-31 M=0..15 {K2,K3}
  v2f A[2][6];
#pragma unroll
  for (int m = 0; m < 2; ++m) {
    const int i = m * 16 + ln;
#pragma unroll
    for (int k = 0; k < 6; ++k) {
      const int j0 = k * 4 + (hi ? 2 : 0);
      const float a0 = (i < C_ && j0     < C_) ? compat[i * C_ + j0]     : 0.f;
      const float a1 = (i < C_ && j0 + 1 < C_) ? compat[i * C_ + j0 + 1] : 0.f;
      A[m][k] = (v2f){a0, a1};
    }
  }

  // --- Accumulate Q_compat = compat x Qs over K=24 (6 steps of 4) ---
  v8f acc0 = {0.f,0.f,0.f,0.f,0.f,0.f,0.f,0.f};
  v8f acc1 = {0.f,0.f,0.f,0.f,0.f,0.f,0.f,0.f};
#pragma unroll
  for (int k = 0; k < 6; ++k) {
    const int j0 = k * 4 + (hi ? 2 : 0);
    const float b0 = (j0     < C_) ? qsB[(size_t)j0       * HW_ + pix] : 0.f;
    const float b1 = (j0 + 1 < C_) ? qsB[(size_t)(j0 + 1) * HW_ + pix] : 0.f;
    const v2f bv = (v2f){b0, b1};
    acc0 = __builtin_amdgcn_wmma_f32_16x16x4_f32(false, A[0][k], false, bv,
                                                 (short)0, acc0, false, false);
    acc1 = __builtin_amdgcn_wmma_f32_16x16x4_f32(false, A[1][k], false, bv,
                                                 (short)0, acc1, false, false);
  }

  // --- x = Q - Qcompat, softmax over 21 classes per pixel ---
  // D layout: VGPR r, lane<16 -> class (mt*16 + r), lane>=16 -> class (mt*16+r+8)
  float e0[8], e1[5];
  float m = -3.4e38f;
#pragma unroll
  for (int r = 0; r < 8; ++r) {
    const int cls = r + (hi ? 8 : 0);        // classes 0..15 across halves
    const float x = qB[(size_t)cls * HW_ + pix] - acc0[r];
    e0[r] = x;
    m = fmaxf(m, x);
  }
#pragma unroll
  for (int r = 0; r < 5; ++r) {              // classes 16..20 live in lanes<16
    const float x = (!hi) ? (qB[(size_t)(16 + r) * HW_ + pix] - acc1[r])
                          : -3.4e38f;
    e1[r] = x;
    if (!hi) m = fmaxf(m, x);
  }
  m = fmaxf(m, __shfl_xor(m, 16, 32));       // pair-reduce across lane halves

  float s = 0.f;
#pragma unroll
  for (int r = 0; r < 8; ++r) { e0[r] = __expf(e0[r] - m); s += e0[r]; }
#pragma unroll
  for (int r = 0; r < 5; ++r) {
    const float t = __expf(e1[r] - m);       // hi lanes: exp(-huge)=0, harmless
    e1[r] = t;
    if (!hi) s += t;
  }
  s += __shfl_xor(s, 16, 32);
  const float inv = 1.f / s;

#pragma unroll
  for (int r = 0; r < 8; ++r) {
    const int cls = r + (hi ? 8 : 0);
    qB[(size_t)cls * HW_ + pix] = e0[r] * inv;
  }
#pragma unroll
  for (int r = 0; r < 5; ++r) {
    if (!hi) qB[(size_t)(16 + r) * HW_ + pix] = e1[r] * inv;
  }
}

// ---------------------------------------------------------------------------
extern "C" void kernel_launch(void* const* d_in, const int* in_sizes, int n_in,
                              void* d_out, int out_size, void* d_ws, size_t ws_size,
                              hipStream_t stream) {
  const float* unary  = (const float*)d_in[0];
  // d_in[1] = image: unused by the faithful reference math
  const float* pos_w  = (const float*)d_in[2];
  const float* bi_w   = (const float*)d_in[3];
  const float* compat = (const float*)d_in[4];

  float* q  = (float*)d_out;   // Q lives in the output buffer (updated in place)
  float* qs = (float*)d_ws;    // blurred+scaled Q, B*C*HW floats (~176 MB)

  // Q0 = softmax(unary)
  crf_softmax_init<<<(B_ * HW_) / 256, 256, 0, stream>>>(unary, q);

  const dim3 blurGrid(W_ / 32, H_ / 32, B_ * C_);
  const int cmpBlocks = (B_ * HW_ / 16) * 32 / 256;   // 16384, exact coverage

  for (int it = 0; it < 5; ++it) {
    crf_blur<<<blurGrid, 256, 0, stream>>>(q, qs, pos_w, bi_w);
    crf_compat_softmax<<<cmpBlocks, 256, 0, stream>>>(q, qs, compat);
  }
}